// MnnRnn_89635967467898
// MI455X (gfx1250) — compile-verified
//
#include <hip/hip_runtime.h>
#include <math.h>

#define N_DIM   512
#define IN_DIM  85
#define OUT_DIM 33
#define T_STP   32
#define BATCH   4

constexpr float ALPHA_C     = 20.0f / 100.0f;          // DT/TAU
constexpr float SIGMA_REC   = 0.05f;
constexpr float SIGMA_INPUT = 0.1f;
constexpr float T_REF_C     = 5.0f;
constexpr float SCALE_C     = 1.0f;
constexpr float BIAS_MU_C   = 0.0f;
constexpr float VAR_S       = (2.0f / ALPHA_C) * SIGMA_REC * SIGMA_REC; // 0.025

typedef __attribute__((ext_vector_type(2))) float v2f;
typedef __attribute__((ext_vector_type(8))) float v8f;
typedef __attribute__((ext_vector_type(4))) int   v4i;

// ---- CDNA5 async global->LDS path (ASYNCcnt-tracked), with sync fallback ----
#if defined(__has_builtin)
#  if __has_builtin(__builtin_amdgcn_global_load_async_to_lds_b128)
#    define ASYNC_LDS 1
#  endif
#endif

#ifdef ASYNC_LDS
#  if __has_builtin(__builtin_amdgcn_s_wait_asynccnt)
#    define WAITASYNC(n) __builtin_amdgcn_s_wait_asynccnt(n)
#  else
#    define WAITASYNC(n) asm volatile("s_wait_asynccnt " #n ::: "memory")
#  endif
#else
#  define WAITASYNC(n) do { } while (0)
#endif

typedef __attribute__((address_space(1))) v4i gv4i;   // global 16B vector
typedef __attribute__((address_space(3))) v4i lv4i;   // LDS 16B vector

__device__ __forceinline__ void async_copy16(const float* g, float* l)
{
#ifdef ASYNC_LDS
    __builtin_amdgcn_global_load_async_to_lds_b128(
        (gv4i*)const_cast<float*>(g),
        (lv4i*)l,
        0, 0);
#else
    float4 v = *(const float4*)g;
    *(float4*)l = v;
#endif
}

// ---------------------------------------------------------------------------
// Batched 512x512x512 fp32 GEMM using V_WMMA_F32_16X16X4_F32.
//   D[b] = A[b] * B[b]           (BT=false)
//   D[b] = A[b] * B[b]^T (+ epi) (BT=true)
// EPILOGUE adds cov_input + var_s * I (stage 2 of W cov W^T).
// Block: 128 threads (4 waves), 64x64 tile; wave -> 32x32 (2x2 WMMA tiles).
// Double-buffered LDS panels filled by GLOBAL_LOAD_ASYNC_TO_LDS_B128;
// s_wait_asynccnt<=4 retires the previous chunk (async loads complete in
// order; each thread issues exactly 4 B128 transfers per chunk).
// ---------------------------------------------------------------------------
template <bool BT, bool EPILOGUE>
__global__ __launch_bounds__(128) void gemm512_wmma(
    const float* __restrict__ A, long aBatchStride,
    const float* __restrict__ B, long bBatchStride,
    float* __restrict__ D, const float* __restrict__ covInput)
{
    constexpr int LBSZ = BT ? (64 * 17) : (16 * 65);
    __shared__ float lA[2][64 * 17];     // A panel: 64 rows x 16 k, pad 17
    __shared__ float lB[2][LBSZ];        // BT: W panel [64][17]; NN: [16][65]

    const int b = blockIdx.z;
    A += (long)b * aBatchStride;
    B += (long)b * bBatchStride;
    D += (long)b * (long)(N_DIM * N_DIM);

    const int tid     = threadIdx.x;
    const int wave    = tid >> 5;
    const int lane    = tid & 31;
    const int laneLow = lane & 15;
    const int hi      = lane >> 4;          // half-wave: K offset 0 or 2
    const int mBlock  = blockIdx.y * 64;
    const int nBlock  = blockIdx.x * 64;
    const int mWave   = (wave >> 1) * 32;
    const int nWave   = (wave & 1) * 32;

    auto issue_chunk = [&](int k0, int buf) {
        // A panel: 64 rows x 16 k (16B-contiguous sub-rows, pad preserved)
#pragma unroll
        for (int i = 0; i < 2; ++i) {
            int t = tid + i * 128;
            int r = t >> 2, c4 = (t & 3) * 4;
            async_copy16(&A[(long)(mBlock + r) * N_DIM + k0 + c4],
                         &lA[buf][r * 17 + c4]);
        }
        if (BT) {
            // keep W untransposed: rows n=0..63, cols k (pad 17)
#pragma unroll
            for (int i = 0; i < 2; ++i) {
                int t = tid + i * 128;
                int r = t >> 2, c4 = (t & 3) * 4;
                async_copy16(&B[(long)(nBlock + r) * N_DIM + k0 + c4],
                             &lB[buf][r * 17 + c4]);
            }
        } else {
            // B panel: 16 k rows x 64 cols (pad 65)
#pragma unroll
            for (int i = 0; i < 2; ++i) {
                int t = tid + i * 128;
                int k = t >> 4, n4 = (t & 15) * 4;
                async_copy16(&B[(long)(k0 + k) * N_DIM + nBlock + n4],
                             &lB[buf][k * 65 + n4]);
            }
        }
    };

    v8f acc[2][2] = {};

    issue_chunk(0, 0);
    constexpr int NCHUNK = N_DIM / 16;                      // 32
    for (int c = 0; c < NCHUNK; ++c) {
        if (c + 1 < NCHUNK) {
            issue_chunk((c + 1) * 16, (c + 1) & 1);
            WAITASYNC(4);                                   // chunk c retired
        } else {
            WAITASYNC(0);
        }
        __syncthreads();

        const float* cA = lA[c & 1];
        const float* cB = lB[c & 1];
#pragma unroll
        for (int kk = 0; kk < 16; kk += 4) {
            const int ka = kk + hi * 2;  // lanes0-15 K={k,k+1}, lanes16-31 K={k+2,k+3}
            v2f afr[2], bfr[2];
#pragma unroll
            for (int mi = 0; mi < 2; ++mi) {
                int row = mWave + mi * 16 + laneLow;
                afr[mi].x = cA[row * 17 + ka];
                afr[mi].y = cA[row * 17 + ka + 1];
            }
#pragma unroll
            for (int ni = 0; ni < 2; ++ni) {
                int col = nWave + ni * 16 + laneLow;
                if (BT) {               // B_eff[k][n] = W[n][k] read stride-17
                    bfr[ni].x = cB[col * 17 + ka];
                    bfr[ni].y = cB[col * 17 + ka + 1];
                } else {
                    bfr[ni].x = cB[ka * 65 + col];
                    bfr[ni].y = cB[(ka + 1) * 65 + col];
                }
            }
#pragma unroll
            for (int mi = 0; mi < 2; ++mi)
#pragma unroll
                for (int ni = 0; ni < 2; ++ni)
                    acc[mi][ni] = __builtin_amdgcn_wmma_f32_16x16x4_f32(
                        false, afr[mi], false, bfr[ni],
                        (short)0, acc[mi][ni], false, false);
        }
        __syncthreads();
    }

    // epilogue: C/D layout -> VGPR v holds M = base + hi*8 + v, N = laneLow
#pragma unroll
    for (int mi = 0; mi < 2; ++mi) {
#pragma unroll
        for (int ni = 0; ni < 2; ++ni) {
            const int col = nBlock + nWave + ni * 16 + laneLow;
#pragma unroll
            for (int v = 0; v < 8; ++v) {
                const int row = mBlock + mWave + mi * 16 + hi * 8 + v;
                float val = acc[mi][ni][v];
                if (EPILOGUE) {
                    val += covInput[(long)row * N_DIM + col];
                    if (row == col) val += VAR_S;
                }
                D[(long)row * N_DIM + col] = val;
            }
        }
    }
}

// ---------------------------------------------------------------------------
// Small helper kernels
// ---------------------------------------------------------------------------
__global__ void init_cov_kernel(const float* __restrict__ cov0, float* __restrict__ covc)
{
    int idx = blockIdx.x * 256 + threadIdx.x;            // 65536 float4
    float4 v = ((const float4*)cov0)[idx];
#pragma unroll
    for (int b = 0; b < BATCH; ++b)
        ((float4*)covc)[(size_t)b * (N_DIM * N_DIM / 4) + idx] = v;
}

__global__ void init_mu_kernel(const float* __restrict__ muIn, float* __restrict__ mu0)
{
    int idx = blockIdx.x * 256 + threadIdx.x;            // 2048
    mu0[idx] = muIn[idx];
}

__global__ void cov_input_kernel(const float* __restrict__ Win, float* __restrict__ covIn)
{
    int idx = blockIdx.x * 256 + threadIdx.x;            // 512*512
    int i = idx >> 9, j = idx & 511;
    float acc = 0.f;
    for (int d = 0; d < IN_DIM; ++d)
        acc += Win[i * IN_DIM + d] * Win[j * IN_DIM + d];
    covIn[idx] = acc * (SIGMA_INPUT * SIGMA_INPUT);
}

__global__ void in_proj_kernel(const float* __restrict__ x, const float* __restrict__ Win,
                               float* __restrict__ proj)
{
    int idx = blockIdx.x * 256 + threadIdx.x;            // T*B*N = 65536
    int n  = idx & 511;
    int tb = idx >> 9;
    const float* xp = x + tb * IN_DIM;
    float acc = 0.f;
    for (int d = 0; d < IN_DIM; ++d)
        acc += xp[d] * Win[n * IN_DIM + d];
    proj[idx] = acc;
}

// mubar = mu W_rec^T + b_rec + in_proj[t]; phi; update mu; emit chi
__global__ __launch_bounds__(256) void phi_kernel(
    const float* __restrict__ muCur, float* __restrict__ muNxt,
    const float* __restrict__ Wrec, const float* __restrict__ brec,
    const float* __restrict__ projT, const float* __restrict__ Cbar,
    float* __restrict__ chi)
{
    __shared__ float sMu[N_DIM];
    const int gid = blockIdx.x * 256 + threadIdx.x;      // 0..2047, one b per block
    const int b = gid >> 9;
    const int i = gid & 511;
    const float* mub = muCur + b * N_DIM;
    for (int j = threadIdx.x; j < N_DIM; j += 256) sMu[j] = mub[j];
    __syncthreads();

    float acc = 0.f;
    const float* wrow = Wrec + (long)i * N_DIM;
    for (int j = 0; j < N_DIM; ++j) acc += sMu[j] * wrow[j];
    float mubar = acc + brec[i] + projT[gid];

    float diag = Cbar[(long)b * N_DIM * N_DIM + (long)i * N_DIM + i];
    float s    = sqrtf(fmaxf(diag, 0.f));
    float gain = 1.f / (1.f + s);
    float u    = (mubar + BIAS_MU_C) * gain;
    float sig  = 1.f / (1.f + expf(-u));
    float muNew = (SCALE_C / T_REF_C) * sig;
    float chiV  = (SCALE_C / T_REF_C) * sig * (1.f - sig) * gain;

    muNxt[gid] = (1.f - ALPHA_C) * sMu[i] + ALPHA_C * muNew;
    chi[gid]   = chiV;
}

__global__ void outproj_kernel(const float* __restrict__ mu, const float* __restrict__ Wout,
                               float* __restrict__ outT)
{
    int idx = blockIdx.x * blockDim.x + threadIdx.x;     // B*O = 132
    if (idx >= BATCH * OUT_DIM) return;
    int b = idx / OUT_DIM, o = idx % OUT_DIM;
    float acc = 0.f;
    for (int i = 0; i < N_DIM; ++i)
        acc += mu[b * N_DIM + i] * Wout[o * N_DIM + i];
    outT[idx] = 1.f / (1.f + expf(-acc));
}

// cov_c = chi_i chi_j (.) Cbar   (vectorized float4)
__global__ void scale_kernel(const float* __restrict__ Cbar, const float* __restrict__ chi,
                             float* __restrict__ covc)
{
    int idx = blockIdx.x * 256 + threadIdx.x;            // 262144 float4
    int j4 = idx & 127;
    int bi = idx >> 7;                                   // b*512 + i
    int b  = bi >> 9;
    float  ci = chi[bi];
    float4 cj = ((const float4*)(chi + b * N_DIM))[j4];
    float4 v  = ((const float4*)Cbar)[idx];
    v.x *= ci * cj.x; v.y *= ci * cj.y; v.z *= ci * cj.z; v.w *= ci * cj.w;
    ((float4*)covc)[idx] = v;
}

// M[b] = W_out @ cov_f[b]   -> [B, O, N]
__global__ void covproj1_kernel(const float* __restrict__ Wout, const float* __restrict__ covf,
                                float* __restrict__ M)
{
    int idx = blockIdx.x * 256 + threadIdx.x;            // B*O*N = 67584
    if (idx >= BATCH * OUT_DIM * N_DIM) return;
    int i  = idx & 511;
    int bo = idx >> 9;                                   // b*O + o
    int b  = bo / OUT_DIM, o = bo % OUT_DIM;
    float acc = 0.f;
    const float* cp = covf + (long)b * N_DIM * N_DIM + i;
    const float* wo = Wout + o * N_DIM;
    for (int j = 0; j < N_DIM; ++j) acc += wo[j] * cp[(long)j * N_DIM];
    M[idx] = acc;
}

// out_cov[b][o][p] = M[b][o] . W_out[p]
__global__ void covproj2_kernel(const float* __restrict__ M, const float* __restrict__ Wout,
                                float* __restrict__ outCov)
{
    int idx = blockIdx.x * 256 + threadIdx.x;            // B*O*O = 4356
    if (idx >= BATCH * OUT_DIM * OUT_DIM) return;
    int p  = idx % OUT_DIM;
    int bo = idx / OUT_DIM;
    float acc = 0.f;
    const float* mp = M + (long)bo * N_DIM;
    const float* wp = Wout + p * N_DIM;
    for (int i = 0; i < N_DIM; ++i) acc += mp[i] * wp[i];
    outCov[idx] = acc;
}

// ---------------------------------------------------------------------------
extern "C" void kernel_launch(void* const* d_in, const int* in_sizes, int n_in,
                              void* d_out, int out_size, void* d_ws, size_t ws_size,
                              hipStream_t stream)
{
    const float* inputs_seq = (const float*)d_in[0];   // [32,4,85]
    const float* mu_in      = (const float*)d_in[1];   // [1,4,512]
    const float* cov_in     = (const float*)d_in[2];   // [1,1,512,512]
    const float* W_rec      = (const float*)d_in[3];   // [512,512]
    const float* b_rec      = (const float*)d_in[4];   // [512]
    const float* W_in       = (const float*)d_in[5];   // [512,85]
    const float* W_out      = (const float*)d_in[6];   // [33,512]
    float* out = (float*)d_out;                        // 4224 + 4356 floats

    const size_t NN = (size_t)N_DIM * N_DIM;
    float* ws       = (float*)d_ws;
    float* covc     = ws; ws += BATCH * NN;
    float* tmp      = ws; ws += BATCH * NN;
    float* Cbar     = ws; ws += BATCH * NN;
    float* covInput = ws; ws += NN;
    float* inProj   = ws; ws += (size_t)T_STP * BATCH * N_DIM;
    float* mu0      = ws; ws += BATCH * N_DIM;
    float* mu1      = ws; ws += BATCH * N_DIM;
    float* chi      = ws; ws += BATCH * N_DIM;
    float* Mbuf     = ws; ws += (size_t)BATCH * OUT_DIM * N_DIM;
    (void)in_sizes; (void)n_in; (void)out_size; (void)ws_size;

    init_cov_kernel<<<256, 256, 0, stream>>>(cov_in, covc);
    init_mu_kernel<<<8, 256, 0, stream>>>(mu_in, mu0);
    cov_input_kernel<<<1024, 256, 0, stream>>>(W_in, covInput);
    in_proj_kernel<<<256, 256, 0, stream>>>(inputs_seq, W_in, inProj);

    dim3 gGrid(8, 8, BATCH), gBlk(128);
    float* mus[2] = {mu0, mu1};
    for (int t = 0; t < T_STP; ++t) {
        // stage 1: tmp[b] = W_rec @ cov_c[b]
        gemm512_wmma<false, false><<<gGrid, gBlk, 0, stream>>>(
            W_rec, 0L, covc, (long)NN, tmp, nullptr);
        // stage 2: Cbar[b] = tmp[b] @ W_rec^T + var_s I + cov_input
        gemm512_wmma<true, true><<<gGrid, gBlk, 0, stream>>>(
            tmp, (long)NN, W_rec, 0L, Cbar, covInput);
        // phi + mu update
        phi_kernel<<<8, 256, 0, stream>>>(
            mus[t & 1], mus[(t + 1) & 1], W_rec, b_rec,
            inProj + (size_t)t * BATCH * N_DIM, Cbar, chi);
        // output projection for this step
        outproj_kernel<<<1, 256, 0, stream>>>(
            mus[(t + 1) & 1], W_out, out + (size_t)t * BATCH * OUT_DIM);
        // cov_c = chi chi^T (.) Cbar
        scale_kernel<<<1024, 256, 0, stream>>>(Cbar, chi, covc);
    }

    covproj1_kernel<<<264, 256, 0, stream>>>(W_out, covc, Mbuf);
    covproj2_kernel<<<18, 256, 0, stream>>>(Mbuf, W_out,
                                            out + (size_t)T_STP * BATCH * OUT_DIM);
}